// TopKModel_22522808500902
// MI455X (gfx1250) — compile-verified
//
#include <hip/hip_runtime.h>

typedef float v2f __attribute__((ext_vector_type(2)));
typedef float v8f __attribute__((ext_vector_type(8)));

#define DIM 128
#define TOPK 8
#define WAVES 8
#define BLOCK 256
#define TOK_PER_WAVE 16
#define TOK_PER_BLOCK (WAVES * TOK_PER_WAVE)
#define WSTRIDE 132   // padded W row stride (floats) -> conflict-free b64 B-frag loads
#define YSTRIDE 132   // padded y row stride (floats) -> conflict-free C stores/scans
#define LDS_W_FLOATS (DIM * WSTRIDE)
#define LDS_Y_FLOATS_PER_WAVE (TOK_PER_WAVE * YSTRIDE)
#define LDS_TOTAL_BYTES ((LDS_W_FLOATS + WAVES * LDS_Y_FLOATS_PER_WAVE) * 4)

__device__ __forceinline__ void insert8(float v, int idx, float tv[TOPK], int ti[TOPK]) {
  if (v <= tv[TOPK - 1]) return;          // early out: cheaper than full CAS chain
  tv[TOPK - 1] = v; ti[TOPK - 1] = idx;
#pragma unroll
  for (int j = TOPK - 1; j > 0; --j) {    // bubble into sorted-descending list
    if (tv[j] > tv[j - 1]) {
      float fv = tv[j]; tv[j] = tv[j - 1]; tv[j - 1] = fv;
      int   it = ti[j]; ti[j] = ti[j - 1]; ti[j - 1] = it;
    }
  }
}

__global__ __launch_bounds__(BLOCK) void topk_linear_kernel(
    const float* __restrict__ x, const float* __restrict__ W,
    float* __restrict__ outV, int* __restrict__ outI, int nTokens) {
  extern __shared__ float smem[];
  float* lW = smem;                       // 128 x 132 f32
  float* lY = smem + LDS_W_FLOATS;        // 8 waves x 16 x 132 f32

  const int tid  = threadIdx.x;
  const int wave = tid >> 5;
  const int lane = tid & 31;
  const int c    = lane & 15;             // M row (A) / N col (B,C) within half-wave
  const int h    = lane >> 4;             // half-wave select (K split)

  // ---- Stage W (row-major [out][in]) into LDS with padded rows ----
#pragma unroll
  for (int e = 0; e < (DIM * DIM) / BLOCK; ++e) {
    int idx = e * BLOCK + tid;
    lW[(idx >> 7) * WSTRIDE + (idx & 127)] = W[idx];
  }
  __syncthreads();

  const int tokBase = blockIdx.x * TOK_PER_BLOCK + wave * TOK_PER_WAVE;
  float* yW = lY + wave * LDS_Y_FLOATS_PER_WAVE;

  // ---- A fragments: 16 tokens x K=128, f32 WMMA layout (ISA 7.12.2) ----
  // lane(0-15):  row=c, K = 4j+0,4j+1 ; lane(16-31): row=c, K = 4j+2,4j+3
  const float* xr = x + (size_t)(tokBase + c) * DIM + 2 * h;
  v2f a[32];
#pragma unroll
  for (int j = 0; j < 32; ++j)
    a[j] = *reinterpret_cast<const v2f*>(xr + 4 * j);

  // ---- GEMM: 8 N-tiles x (K=128 in steps of 4) via V_WMMA_F32_16X16X4_F32 ----
#pragma unroll
  for (int t = 0; t < 8; ++t) {
    // B[k][n] = W[n][k] : lane holds column n = t*16 + c, K striped like A
    const float* wrow = lW + (t * 16 + c) * WSTRIDE + 2 * h;
    v8f acc = {};
#pragma unroll
    for (int j = 0; j < 32; ++j) {
      v2f b = *reinterpret_cast<const v2f*>(wrow + 4 * j);
      acc = __builtin_amdgcn_wmma_f32_16x16x4_f32(
          false, a[j], false, b, (short)0, acc, false, false);
    }
    // C layout: VGPR v -> row v + 8*h, col = t*16 + c
#pragma unroll
    for (int v = 0; v < 8; ++v)
      yW[(v + 8 * h) * YSTRIDE + t * 16 + c] = acc[v];
  }

  // ---- top-8: two lanes per row, each scans 64 columns ----
  float tv[TOPK]; int ti[TOPK];
#pragma unroll
  for (int p = 0; p < TOPK; ++p) { tv[p] = -__builtin_inff(); ti[p] = 0; }

  const int r = c;                        // token row within the wave tile
  for (int i = 0; i < 64; ++i) {
    int col = h * 64 + ((i + 2 * h) & 63);   // +2 rotation: bank-conflict-free
    float v = yW[r * YSTRIDE + col];
    insert8(v, col, tv, ti);
  }

  // ---- merge the two half-lists via LDS scratch (reuse y region; DS in-order) ----
  float* ms = yW;                          // 32 lanes * 16 dwords = 2 KB
#pragma unroll
  for (int p = 0; p < TOPK; ++p) {
    ms[lane * 16 + 2 * p]     = tv[p];
    ms[lane * 16 + 2 * p + 1] = __int_as_float(ti[p]);
  }
  if (h == 0) {
#pragma unroll
    for (int p = 0; p < TOPK; ++p) {
      float v  = ms[(lane + 16) * 16 + 2 * p];
      int   id = __float_as_int(ms[(lane + 16) * 16 + 2 * p + 1]);
      insert8(v, id, tv, ti);
    }
    const size_t token = (size_t)tokBase + r;
    float4 v0 = make_float4(tv[0], tv[1], tv[2], tv[3]);
    float4 v1 = make_float4(tv[4], tv[5], tv[6], tv[7]);
    *reinterpret_cast<float4*>(outV + token * TOPK)     = v0;
    *reinterpret_cast<float4*>(outV + token * TOPK + 4) = v1;
    int4 i0 = make_int4(ti[0], ti[1], ti[2], ti[3]);
    int4 i1 = make_int4(ti[4], ti[5], ti[6], ti[7]);
    *reinterpret_cast<int4*>(outI + token * TOPK)     = i0;
    *reinterpret_cast<int4*>(outI + token * TOPK + 4) = i1;
  }
}

extern "C" void kernel_launch(void* const* d_in, const int* in_sizes, int n_in,
                              void* d_out, int out_size, void* d_ws, size_t ws_size,
                              hipStream_t stream) {
  const float* x = (const float*)d_in[0];
  const float* W = (const float*)d_in[1];
  const int nTokens = in_sizes[0] / DIM;

  float* outV = (float*)d_out;                         // values [N,8] f32
  int*   outI = (int*)d_out + (size_t)nTokens * TOPK;  // indices [N,8] i32 (bit-stored)

  (void)hipFuncSetAttribute((const void*)topk_linear_kernel,
                            hipFuncAttributeMaxDynamicSharedMemorySize,
                            LDS_TOTAL_BYTES);

  const int grid = nTokens / TOK_PER_BLOCK;            // 1048576/128 = 8192
  topk_linear_kernel<<<grid, BLOCK, LDS_TOTAL_BYTES, stream>>>(x, W, outV, outI, nTokens);
}